// GCN_26096221290966
// MI455X (gfx1250) — compile-verified
//
#include <hip/hip_runtime.h>
#include <hip/hip_bf16.h>

typedef __attribute__((ext_vector_type(2))) float v2f;
typedef __attribute__((ext_vector_type(8))) float v8f;

#define N_NODES 100000
#define N_EDGES 1000000
#define C_HID   128

// ---------------------------------------------------------------------------
// Degree / normalization
// ---------------------------------------------------------------------------
__global__ void gcn_init_deg(float* deg, int n) {
    int i = blockIdx.x * blockDim.x + threadIdx.x;
    if (i < n) deg[i] = 1.0f;  // self-loop contributes 1 to in-degree
}

__global__ void gcn_edge_deg(const int* col, float* deg, int e) {
    int i = blockIdx.x * blockDim.x + threadIdx.x;
    if (i < e) atomicAdd(&deg[col[i]], 1.0f);
}

__global__ void gcn_rsqrt(float* deg, int n) {
    int i = blockIdx.x * blockDim.x + threadIdx.x;
    if (i < n) deg[i] = rsqrtf(deg[i]);  // deg >= 1 always (self loops)
}

// ---------------------------------------------------------------------------
// fp32 WMMA GEMM: Y[N,128] = X[N,K] @ W[K,128], K in {64,128}, N % 16 == 0
// Block = 256 threads (8 waves). Each block: one 16-row M tile, each wave one
// 16x16 N tile. K staged through LDS in chunks of 32.
// ---------------------------------------------------------------------------
__global__ void gcn_gemm_wmma(const float* __restrict__ X,
                              const float* __restrict__ W,
                              float* __restrict__ Y, int K) {
    __shared__ float sX[16 * 32];     // 2 KB
    __shared__ float sW[32 * 128];    // 16 KB

    const int tid  = threadIdx.x;
    const int wave = tid >> 5;
    const int lane = tid & 31;
    const int m0   = blockIdx.x * 16;

    v8f acc = {};

    const int m     = lane & 15;
    const int n     = (wave << 4) + (lane & 15);
    const int khalf = (lane >> 4) << 1;   // lanes 0-15 -> K+0/K+1, lanes 16-31 -> K+2/K+3

    for (int kc = 0; kc < K; kc += 32) {
        // stage X tile: 16x32
        for (int i = tid; i < 16 * 32; i += 256) {
            int r = i >> 5, c = i & 31;
            sX[i] = X[(m0 + r) * K + kc + c];
        }
        // stage W tile: 32x128 (float4 per thread x4)
        for (int i = tid * 4; i < 32 * 128; i += 256 * 4) {
            *(float4*)&sW[i] = *(const float4*)&W[(kc + (i >> 7)) * 128 + (i & 127)];
        }
        __syncthreads();

#pragma unroll
        for (int k = 0; k < 32; k += 4) {
            v2f a, b;
            a.x = sX[m * 32 + k + khalf];
            a.y = sX[m * 32 + k + khalf + 1];
            b.x = sW[(k + khalf) * 128 + n];
            b.y = sW[(k + khalf + 1) * 128 + n];
            acc = __builtin_amdgcn_wmma_f32_16x16x4_f32(
                false, a, false, b, (short)0, acc, false, false);
        }
        __syncthreads();
    }

    // C layout: VGPR r -> row r (lanes 0-15) / row r+8 (lanes 16-31), col = lane&15
    const int rowoff = (lane >> 4) << 3;
#pragma unroll
    for (int r = 0; r < 8; ++r) {
        Y[(m0 + r + rowoff) * 128 + n] = acc[r];
    }
}

// ---------------------------------------------------------------------------
// Zero fill (float4)
// ---------------------------------------------------------------------------
__global__ void gcn_zero(float* p, int n4) {
    int i = blockIdx.x * blockDim.x + threadIdx.x;
    if (i < n4) *(float4*)&p[i * 4] = make_float4(0.f, 0.f, 0.f, 0.f);
}

// ---------------------------------------------------------------------------
// Edge scatter: one wave32 per edge, 4 channels per lane.
// out[col] += xw[row] * dinv[row] * dinv[col]
// ---------------------------------------------------------------------------
__global__ void gcn_scatter(const float* __restrict__ xw,
                            const float* __restrict__ dinv,
                            const int* __restrict__ row,
                            const int* __restrict__ col,
                            float* __restrict__ out, int e) {
    int gw   = (blockIdx.x * blockDim.x + threadIdx.x) >> 5;
    int lane = threadIdx.x & 31;
    if (gw >= e) return;
    int r = row[gw];   // lane-uniform -> broadcast load
    int c = col[gw];
    float norm = dinv[r] * dinv[c];
    const float4 v = *(const float4*)&xw[r * 128 + lane * 4];
    float* dst = &out[c * 128 + lane * 4];
    atomicAdd(dst + 0, v.x * norm);
    atomicAdd(dst + 1, v.y * norm);
    atomicAdd(dst + 2, v.z * norm);
    atomicAdd(dst + 3, v.w * norm);
}

// ---------------------------------------------------------------------------
// Finalize: h = relu(accum + xw*dinv^2 (self loop) + bias), in place on accum
// ---------------------------------------------------------------------------
__global__ void gcn_finalize(float* __restrict__ accum,
                             const float* __restrict__ xw,
                             const float* __restrict__ dinv,
                             const float* __restrict__ bias, int total) {
    int idx = blockIdx.x * blockDim.x + threadIdx.x;
    if (idx >= total) return;
    int i = idx >> 7, c = idx & 127;
    float d = dinv[i];
    float v = accum[idx] + xw[idx] * d * d + bias[c];
    accum[idx] = fmaxf(v, 0.0f);
}

// ---------------------------------------------------------------------------
// Graph counts
// ---------------------------------------------------------------------------
__global__ void gcn_count(const int* __restrict__ batch, float* cnt, int n) {
    int i = blockIdx.x * blockDim.x + threadIdx.x;
    if (i < n) atomicAdd(&cnt[batch[i]], 1.0f);
}

// ---------------------------------------------------------------------------
// Pool: LDS-staged segment sum. batch is sorted, so each block of 128 nodes
// spans a small graph range [gmin, gmax]; accumulate in LDS (ds_add_f32),
// flush only that range with global atomics.
// ---------------------------------------------------------------------------
#define POOL_NODES 128
__global__ void gcn_pool(const float* __restrict__ h,
                         const int* __restrict__ batch,
                         float* __restrict__ pooled, int n) {
    __shared__ float sP[64 * 128];   // 32 KB
    __shared__ int sGmin, sGmax;
    int tid = threadIdx.x;
    for (int i = tid; i < 64 * 128; i += 256) sP[i] = 0.0f;

    int n0 = blockIdx.x * POOL_NODES;
    int n1 = n0 + POOL_NODES;
    if (n1 > n) n1 = n;
    if (tid == 0) { sGmin = batch[n0]; sGmax = batch[n1 - 1]; }
    __syncthreads();

    int c = tid & 127;
    for (int nd = n0 + (tid >> 7); nd < n1; nd += 2) {
        atomicAdd(&sP[batch[nd] * 128 + c], h[nd * 128 + c]);
    }
    __syncthreads();

    int gmin = sGmin, gmax = sGmax;
    int span = (gmax - gmin + 1) * 128;
    for (int i = tid; i < span; i += 256) {
        atomicAdd(&pooled[gmin * 128 + i], sP[gmin * 128 + i]);
    }
}

// ---------------------------------------------------------------------------
// FC: out[g,o] = relu((sum_k pooled_sum[g,k] * Wfc[k,o]) / cnt[g] + bfc[o])
// ---------------------------------------------------------------------------
__global__ void gcn_fc(const float* __restrict__ pooled,
                       const float* __restrict__ cnt,
                       const float* __restrict__ Wfc,
                       const float* __restrict__ bfc,
                       float* __restrict__ out) {
    int idx = blockIdx.x * blockDim.x + threadIdx.x;  // 64*128
    if (idx >= 64 * 128) return;
    int g = idx >> 7, o = idx & 127;
    float s = 0.0f;
#pragma unroll 4
    for (int k = 0; k < 128; ++k) s += pooled[g * 128 + k] * Wfc[k * 128 + o];
    float cc = fmaxf(cnt[g], 1.0f);
    out[idx] = fmaxf(s / cc + bfc[o], 0.0f);
}

// ---------------------------------------------------------------------------
// Launch
// ---------------------------------------------------------------------------
extern "C" void kernel_launch(void* const* d_in, const int* in_sizes, int n_in,
                              void* d_out, int out_size, void* d_ws, size_t ws_size,
                              hipStream_t stream) {
    const float* x    = (const float*)d_in[0];
    const int*   ei   = (const int*)d_in[1];   // [2, E]: row = ei, col = ei + E
    const int*   batch= (const int*)d_in[2];
    const float* W1   = (const float*)d_in[3];
    const float* b1   = (const float*)d_in[4];
    const float* W2   = (const float*)d_in[5];
    const float* b2   = (const float*)d_in[6];
    const float* Wfc  = (const float*)d_in[7];
    const float* bfc  = (const float*)d_in[8];
    float* out = (float*)d_out;

    const int N = N_NODES, E = N_EDGES;
    const int* row = ei;
    const int* col = ei + E;

    float* ws     = (float*)d_ws;
    float* dinv   = ws;                       // N floats
    float* bufA   = dinv + N;                 // N*128 (xw)
    float* bufB   = bufA + (size_t)N * 128;   // N*128 (accum / h)
    float* pooled = bufB + (size_t)N * 128;   // 64*128
    float* cnt    = pooled + 64 * 128;        // 64

    dim3 blk(256);

    // degree normalization
    gcn_init_deg<<<(N + 255) / 256, blk, 0, stream>>>(dinv, N);
    gcn_edge_deg<<<(E + 255) / 256, blk, 0, stream>>>(col, dinv, E);
    gcn_rsqrt<<<(N + 255) / 256, blk, 0, stream>>>(dinv, N);

    const int mtiles = N / 16;                       // 6250 exactly
    const int scat_blocks = (E * 32 + 255) / 256;    // one wave per edge
    const int tot = N * 128;

    // layer 1
    gcn_gemm_wmma<<<mtiles, blk, 0, stream>>>(x, W1, bufA, 64);
    gcn_zero<<<(tot / 4 + 255) / 256, blk, 0, stream>>>(bufB, tot / 4);
    gcn_scatter<<<scat_blocks, blk, 0, stream>>>(bufA, dinv, row, col, bufB, E);
    gcn_finalize<<<(tot + 255) / 256, blk, 0, stream>>>(bufB, bufA, dinv, b1, tot);

    // layer 2
    gcn_gemm_wmma<<<mtiles, blk, 0, stream>>>(bufB, W2, bufA, 128);
    gcn_zero<<<(tot / 4 + 255) / 256, blk, 0, stream>>>(bufB, tot / 4);
    gcn_scatter<<<scat_blocks, blk, 0, stream>>>(bufA, dinv, row, col, bufB, E);
    gcn_finalize<<<(tot + 255) / 256, blk, 0, stream>>>(bufB, bufA, dinv, b2, tot);

    // mean pool + fc
    gcn_zero<<<((64 * 128 + 64) / 4 + 255) / 256, blk, 0, stream>>>(pooled, (64 * 128 + 64) / 4);
    gcn_count<<<(N + 255) / 256, blk, 0, stream>>>(batch, cnt, N);
    gcn_pool<<<(N + POOL_NODES - 1) / POOL_NODES, blk, 0, stream>>>(bufB, batch, pooled, N);
    gcn_fc<<<(64 * 128 + 255) / 256, blk, 0, stream>>>(pooled, cnt, Wfc, bfc, out);
}